// TNRDStage_73538430043016
// MI455X (gfx1250) — compile-verified
//
#include <hip/hip_runtime.h>
#include <hip/hip_bf16.h>
#include <math.h>

typedef float v2f __attribute__((ext_vector_type(2)));
typedef float v8f __attribute__((ext_vector_type(8)));

#define IMG_H 512
#define IMG_W 512
#define IMG_B 8
#define NFILT 24
#define NBASIS 31
#define TW 32          // output tile width
#define TH 16          // output tile height
#define PW (TW + 4)    // phi region width  (36)
#define PH (TH + 4)    // phi region height (20)
#define UW (TW + 8)    // u tile width  (40)
#define UH (TH + 8)    // u tile height (24)
#define NSEG ((PW * PH) / 16)  // 45 sixteen-pixel GEMM segments
#define RED_BLOCKS 1024

__device__ __forceinline__ float clip01(float v, float lo, float hi) {
    return fminf(fmaxf(v, lo), hi);
}

// ---------- Stage 1: per-block partial sums of sum(avg_pool3(clip(u))) ----------
// sum_p u_sigma[p] == sum_q clip(u[q]) * cy(q)*cx(q)/9 with cy,cx in {2,3}.
__global__ __launch_bounds__(256) void tnrd_reduce1(const float* __restrict__ u,
                                                    float* __restrict__ partials) {
    __shared__ float sdata[256];
    const int total = IMG_B * IMG_H * IMG_W;
    float s = 0.f;
    for (int idx = blockIdx.x * 256 + threadIdx.x; idx < total; idx += 256 * RED_BLOCKS) {
        int x = idx & (IMG_W - 1);
        int y = (idx >> 9) & (IMG_H - 1);
        float v = clip01(u[idx], 1e-4f, 1.0f);
        float cy = (y == 0 || y == IMG_H - 1) ? 2.f : 3.f;
        float cx = (x == 0 || x == IMG_W - 1) ? 2.f : 3.f;
        s += v * cy * cx * (1.f / 9.f);
    }
    sdata[threadIdx.x] = s;
    __syncthreads();
    for (int off = 128; off > 0; off >>= 1) {
        if (threadIdx.x < off) sdata[threadIdx.x] += sdata[threadIdx.x + off];
        __syncthreads();
    }
    if (threadIdx.x == 0) partials[blockIdx.x] = sdata[0];
}

// ---------- Stage 2: deterministic final reduction ----------
__global__ __launch_bounds__(256) void tnrd_reduce2(const float* __restrict__ partials,
                                                    float* __restrict__ Sout) {
    __shared__ float sdata[256];
    float s = 0.f;
    for (int i = threadIdx.x; i < RED_BLOCKS; i += 256) s += partials[i];
    sdata[threadIdx.x] = s;
    __syncthreads();
    for (int off = 128; off > 0; off >>= 1) {
        if (threadIdx.x < off) sdata[threadIdx.x] += sdata[threadIdx.x + off];
        __syncthreads();
    }
    if (threadIdx.x == 0) Sout[0] = sdata[0];
}

// ---------- Stage 3: fully fused conv1(WMMA) + RBF + scale + conv2 + reaction ----------
__global__ __launch_bounds__(256) void tnrd_fused(const float* __restrict__ u,
                                                  const float* __restrict__ f,
                                                  const float* __restrict__ filt,
                                                  const float* __restrict__ mu,
                                                  const float* __restrict__ wgt,
                                                  const float* __restrict__ lam,
                                                  const float* __restrict__ Sptr,
                                                  float* __restrict__ out) {
    __shared__ float    u_lds[UH * UW];              //  3840 B (f32, feeds WMMA A)
    __shared__ _Float16 phi_lds[NFILT * PH * PW];    // 34560 B (f16 scaled_phi)
    __shared__ float    w_lds[NFILT * NBASIS];       //  2976 B
    __shared__ float    mu_lds[NFILT * NBASIS];      //  2976 B
                                                     // total ~44.3 KB

    const int tid = threadIdx.x;
    const int b  = blockIdx.z;
    const int oy = blockIdx.y * TH;
    const int ox = blockIdx.x * TW;

    const float S = Sptr[0];
    const float invM = 1.0f / (S / (float)(IMG_B * IMG_H * IMG_W) + 0.001f);
    const float lambda_p = lam[0];

    // prefetch the f tile (only consumed at the very end of the kernel)
    {
        const size_t base = (size_t)b * IMG_H * IMG_W + (size_t)oy * IMG_W + ox;
        __builtin_prefetch(f + base + (size_t)(tid & 15) * IMG_W + (tid >> 4), 0, 0);
    }

    // ---- stage clipped u tile (zero outside image == conv zero padding) ----
    for (int i = tid; i < UH * UW; i += 256) {
        int uy = i / UW, ux = i - uy * UW;
        int gy = oy - 4 + uy, gx = ox - 4 + ux;
        float v = 0.f;
        if (gy >= 0 && gy < IMG_H && gx >= 0 && gx < IMG_W)
            v = clip01(u[(size_t)b * IMG_H * IMG_W + gy * IMG_W + gx], 1e-4f, 1.0f);
        u_lds[i] = v;
    }
    for (int i = tid; i < NFILT * NBASIS; i += 256) {
        w_lds[i]  = wgt[i];
        mu_lds[i] = mu[i];
    }
    __syncthreads();

    const int lane = tid & 31;
    const int half = lane >> 4;   // lane half selects K parity block per ISA layout
    const int nl   = lane & 15;
    const int wave = tid >> 5;

    // ---- preload B matrices (filters as [K=25pad28, N=24pad32]) into registers ----
    // V_WMMA_F32_16X16X4_F32 B layout: VGPR v, lane-half h holds row K = 4*kk + 2h + v
    float bB[2][7][2];
#pragma unroll
    for (int c = 0; c < 2; ++c)
#pragma unroll
        for (int kk = 0; kk < 7; ++kk)
#pragma unroll
            for (int v = 0; v < 2; ++v) {
                int n = c * 16 + nl;
                int k = kk * 4 + 2 * half + v;
                bB[c][kk][v] = (n < NFILT && k < 25) ? filt[n * 25 + k] : 0.f;
            }

    // ---- conv1 GEMM over 45 segments of 16 phi-region pixels, 8 waves round-robin ----
    for (int seg = wave; seg < NSEG; seg += 8) {
        // A layout: lane m=nl is the pixel row; VGPR v, half h holds K = 4*kk + 2h + v
        const int piA = seg * 16 + nl;
        const int eyA = piA / PW, exA = piA - eyA * PW;
        float A[7][2];
#pragma unroll
        for (int kk = 0; kk < 7; ++kk)
#pragma unroll
            for (int v = 0; v < 2; ++v) {
                int k = kk * 4 + 2 * half + v;
                A[kk][v] = (k < 25) ? u_lds[(eyA + k / 5) * UW + exA + (k % 5)] : 0.f;
            }

        // u_sigma/M for this lane's 8 accumulator pixels (C layout: m = 8*half + r)
        float usig[8];
        int inimg[8];
#pragma unroll
        for (int r = 0; r < 8; ++r) {
            int pm = seg * 16 + 8 * half + r;
            int sy = pm / PW, sx = pm - sy * PW;
            float s = 0.f;
#pragma unroll
            for (int iy = 0; iy < 3; ++iy)
#pragma unroll
                for (int ix = 0; ix < 3; ++ix)
                    s += u_lds[(sy + 1 + iy) * UW + (sx + 1 + ix)];
            usig[r] = s * (1.f / 9.f) * invM;
            int py = oy - 2 + sy, px = ox - 2 + sx;
            inimg[r] = (py >= 0 && py < IMG_H && px >= 0 && px < IMG_W);
        }

#pragma unroll
        for (int c = 0; c < 2; ++c) {
            v8f acc = {0.f, 0.f, 0.f, 0.f, 0.f, 0.f, 0.f, 0.f};
#pragma unroll
            for (int kk = 0; kk < 7; ++kk) {
                v2f av = {A[kk][0], A[kk][1]};
                v2f bv = {bB[c][kk][0], bB[c][kk][1]};
                acc = __builtin_amdgcn_wmma_f32_16x16x4_f32(
                    false, av, false, bv, (short)0, acc, false, false);
            }
            const int n = c * 16 + nl;
            if (n < NFILT) {
                // per-lane filter row of RBF params held in registers
                float wr[NBASIS], mr[NBASIS];
#pragma unroll
                for (int j = 0; j < NBASIS; ++j) {
                    wr[j] = w_lds[n * NBASIS + j];
                    mr[j] = mu_lds[n * NBASIS + j];
                }
#pragma unroll
                for (int r = 0; r < 8; ++r) {
                    float cv = acc[r];
                    float phi = 0.f;
#pragma unroll
                    for (int j = 0; j < NBASIS; ++j) {
                        float d = cv - mr[j];
                        // inv(2*gamma^2) = 2.0 for gamma = 0.5
                        phi += wr[j] * __expf(-(d * d) * 2.0f);
                    }
                    int pm = seg * 16 + 8 * half + r;
                    int sy = pm / PW, sx = pm - sy * PW;
                    // zero outside image: conv2 pads scaled_phi with zeros
                    float pv = inimg[r] ? phi * usig[r] : 0.f;
                    phi_lds[(n * PH + sy) * PW + sx] = (_Float16)pv;
                }
            }
        }
    }
    __syncthreads();

    // ---- conv2 (flipped-filter correlation over 24 channels) + reaction + clip ----
    // 512 output pixels, 256 threads: 2 pixels/thread sharing every (uniform,
    // scalar-loaded) filter coefficient.
    {
        const int ty0 = tid / TW, tx = tid - ty0 * TW;   // pixel 0: rows 0..7
        const int ty1 = ty0 + TH / 2;                    // pixel 1: rows 8..15
        float acc0 = 0.f, acc1 = 0.f;
        for (int c = 0; c < NFILT; ++c) {
#pragma unroll
            for (int i = 0; i < 5; ++i)
#pragma unroll
                for (int j = 0; j < 5; ++j) {
                    // uniform index into global -> s_load through constant cache
                    float coef = filt[c * 25 + (4 - i) * 5 + (4 - j)];
                    acc0 += coef * (float)phi_lds[(c * PH + ty0 + i) * PW + (tx + j)];
                    acc1 += coef * (float)phi_lds[(c * PH + ty1 + i) * PW + (tx + j)];
                }
        }
#pragma unroll
        for (int p = 0; p < 2; ++p) {
            const int gy = oy + (p == 0 ? ty0 : ty1);
            const int gx = ox + tx;
            const float acc = (p == 0) ? acc0 : acc1;
            const size_t gi = (size_t)b * IMG_H * IMG_W + (size_t)gy * IMG_W + gx;
            float uv = clip01(u[gi], 1e-4f, 1.0f);
            float fv = clip01(f[gi], 1e-4f, 1.0f);
            float reac = lambda_p * (uv - fv) / (uv + 0.05f);
            out[gi] = clip01(uv - acc - reac, 0.f, 1.f);
        }
    }
}

extern "C" void kernel_launch(void* const* d_in, const int* in_sizes, int n_in,
                              void* d_out, int out_size, void* d_ws, size_t ws_size,
                              hipStream_t stream) {
    (void)in_sizes; (void)n_in; (void)out_size; (void)ws_size;
    const float* u    = (const float*)d_in[0];
    const float* f    = (const float*)d_in[1];
    const float* filt = (const float*)d_in[2];
    const float* mu   = (const float*)d_in[3];
    const float* wgt  = (const float*)d_in[4];
    const float* lam  = (const float*)d_in[5];
    float* out = (float*)d_out;
    float* ws  = (float*)d_ws;

    float* Sv       = ws;      // ws[0]           : final sum
    float* partials = ws + 1;  // ws[1..1024]     : per-block partials

    tnrd_reduce1<<<RED_BLOCKS, 256, 0, stream>>>(u, partials);
    tnrd_reduce2<<<1, 256, 0, stream>>>(partials, Sv);

    dim3 grid(IMG_W / TW, IMG_H / TH, IMG_B);
    tnrd_fused<<<grid, 256, 0, stream>>>(u, f, filt, mu, wgt, lam, Sv, out);
}